// Rank_CLS_Loss_13967233647031
// MI455X (gfx1250) — compile-verified
//
#include <hip/hip_runtime.h>
#include <hip/hip_bf16.h>

// ---------------------------------------------------------------------------
// Rank_CLS_Loss on MI455X (gfx1250).
//
// Bandwidth-bound: 128 MB inputs, fits in 192 MB L2. Plan:
//   P1  stats + radix-histogram (bits 31:21)  [HBM pass, async->LDS staged]
//   S1  per-row scan -> coarse bucket of k-th largest negative
//   P2  refine histogram (bits 20:10)         [L2 pass]
//   S2  per-row scan
//   P3  refine histogram (bits  9:0)          [L2 pass] -> exact 32-bit threshold
//   S3  per-row finalize threshold + tie count (exact, matches sort semantics)
//   P4  weighted pass: S = sum exp(x-M), T = sum x*exp(x-M) over x > thr,
//       pred tile staged via global_load_async_to_lds_b128 (+ s_wait_asynccnt)
//   F   per-row loss + valid-mean -> d_out[0]
//
// WMMA is structurally inapplicable (no matmul); the gfx1250-specific path
// used is the ASYNCcnt-tracked async global->LDS copy engine, overlapped with
// register-side label loads in the two heavy passes.
// ---------------------------------------------------------------------------

constexpr int   B_ROWS = 128;
constexpr int   N_COLS = 131072;
constexpr int   TPB    = 256;          // 8 waves (wave32)
constexpr int   BPR    = 64;           // blocks per row
constexpr int   CHUNK  = N_COLS / BPR; // 2048 elements / block
constexpr int   NGRAN  = CHUNK * 4 / 16; // 512 x 16B granules per tile
constexpr int   GPT    = NGRAN / TPB;    // 2 granules per thread
constexpr int   NB12   = 2048;         // 11-bit buckets (rounds 1,2)
constexpr int   NB3    = 1024;         // 10-bit buckets (round 3)
constexpr float LOSS_L = 4.0f;
constexpr float MARGIN = 0.5f;
constexpr float THS    = 0.5f;
constexpr int   NUM_HARD = 8;

struct Row {
  int      num_pos;
  int      num_neg;
  int      hard;
  float    sum_pos;
  unsigned neg_max_bits;   // atomicMax on bit pattern (values >= 0 -> monotone)
  int      k_eff;
  int      done;
  int      remaining;      // how many still needed from current bucket
  int      r_eq;           // copies of the exact threshold value to include
  unsigned prefix;         // selected high bits so far
  float    v_thr;
  float    S;              // sum exp(x - M) over x > v_thr
  float    T;              // sum x*exp(x - M) over x > v_thr
  int      pad[3];
};
static_assert(sizeof(Row) == 64, "Row must be 64B");

// Issue an async global->LDS copy of one 8 KB pred tile (per-lane 16B, GV
// mode). LDS byte address = low 32 bits of the generic pointer (ISA 10.2:
// the LDS aperture truncates generic addresses to addr[31:0], wave-relative).
__device__ __forceinline__ void async_copy_tile(const float* __restrict__ gp,
                                                float* __restrict__ sp) {
  const unsigned l0  = (unsigned)(size_t)sp;
  const int      tid = threadIdx.x;
#pragma unroll
  for (int j = 0; j < GPT; ++j) {
    const int idx = j * TPB + tid;               // 0..511
    asm volatile("global_load_async_to_lds_b128 %0, %1, off"
                 :: "v"(l0 + idx * 16), "v"(gp + idx * 4) : "memory");
  }
}

__device__ __forceinline__ void async_wait_all() {
  asm volatile("s_wait_asynccnt 0x0" ::: "memory");
}

// -------------------- Pass 1: stats + coarse histogram ---------------------
__global__ void k_stats_hist1(const float* __restrict__ pred,
                              const int*  __restrict__ label,
                              Row* __restrict__ st, int* __restrict__ hist) {
  __shared__ int      lh[NB12];
  __shared__ float    spm[CHUNK];      // async-staged pred tile (8 KB)
  __shared__ int      s_np, s_nn, s_nh;
  __shared__ float    s_sp;
  __shared__ unsigned s_mx;
  const int row = blockIdx.y, blk = blockIdx.x, tid = threadIdx.x;

  const size_t base = (size_t)row * N_COLS + (size_t)blk * CHUNK;

  // 1) kick off the async pred copy first (hides HBM latency behind setup)
  async_copy_tile(pred + base, spm);

  // 2) overlap: zero the LDS histogram + fetch labels into registers
  for (int i = tid; i < NB12; i += TPB) lh[i] = 0;
  if (tid == 0) { s_np = 0; s_nn = 0; s_nh = 0; s_sp = 0.f; s_mx = 0u; }
  const int4* l4 = (const int4*)(label + base);
  int4 lb[GPT];
#pragma unroll
  for (int j = 0; j < GPT; ++j) lb[j] = l4[j * TPB + tid];

  // 3) drain the async engine, then block barrier
  async_wait_all();
  __syncthreads();

  int np = 0, nn = 0, nh = 0; float sp = 0.f; unsigned mx = 0u;
#pragma unroll
  for (int j = 0; j < GPT; ++j) {
    const int e = (j * TPB + tid) * 4;          // lane reads its own granule
    int ls[4] = {lb[j].x, lb[j].y, lb[j].z, lb[j].w};
#pragma unroll
    for (int c = 0; c < 4; ++c) {
      float x = spm[e + c]; int la = ls[c];
      if (la == 1) { ++np; sp += x; }
      else if (la == 0) {
        ++nn;
        if (x > THS) ++nh;
        unsigned kb = __float_as_uint(x);
        mx = kb > mx ? kb : mx;
        atomicAdd(&lh[kb >> 21], 1);            // ds_add_u32
      }
    }
  }
  atomicAdd(&s_np, np); atomicAdd(&s_nn, nn); atomicAdd(&s_nh, nh);
  atomicAdd(&s_sp, sp); atomicMax(&s_mx, mx);
  __syncthreads();

  int* gh = hist + row * NB12;
  for (int i = tid; i < NB12; i += TPB) {
    int c = lh[i];
    if (c) atomicAdd(&gh[i], c);
  }
  if (tid == 0) {
    atomicAdd(&st[row].num_pos, s_np);
    atomicAdd(&st[row].num_neg, s_nn);
    atomicAdd(&st[row].hard,    s_nh);
    atomicAdd(&st[row].sum_pos, s_sp);
    atomicMax(&st[row].neg_max_bits, s_mx);
  }
}

// -------------------- Scan 1: pick coarse bucket ---------------------------
__global__ void k_scan1(Row* __restrict__ st, int* __restrict__ hist) {
  const int row = threadIdx.x;
  if (row >= B_ROWS) return;
  Row& r  = st[row];
  int* h  = hist + row * NB12;
  const int np = r.num_pos, nn = r.num_neg, nh = r.hard;
  int k    = np > 0 ? np : (nh > NUM_HARD ? nh : NUM_HARD);
  int keff = k < nn ? k : nn;                 // -BIG slots contribute 0 exactly
  r.k_eff = keff;
  if (keff <= 0)        { r.done = 1; r.v_thr =  2.0f; r.r_eq = 0; }
  else if (keff >= nn)  { r.done = 1; r.v_thr = -1.0f; r.r_eq = 0; } // take all
  else {
    int cum = 0, b = NB12 - 1;
    for (; b > 0; --b) { int c = h[b]; if (cum + c >= keff) break; cum += c; }
    r.prefix = (unsigned)b; r.remaining = keff - cum; r.done = 0;
  }
  for (int i = 0; i < NB12; ++i) h[i] = 0;    // reuse for next round
}

// -------------------- Passes 2/3: refinement histograms --------------------
template <int ROUND>
__global__ void k_refine(const float* __restrict__ pred,
                         const int*  __restrict__ label,
                         Row* __restrict__ st, int* __restrict__ hist) {
  __shared__ int      lh[NB12];
  __shared__ int      s_done;
  __shared__ unsigned s_pref;
  const int row = blockIdx.y, blk = blockIdx.x, tid = threadIdx.x;
  if (tid == 0) { s_done = st[row].done; s_pref = st[row].prefix; }
  __syncthreads();
  if (s_done) return;                          // block-uniform exit

  const int nb = (ROUND == 2) ? NB12 : NB3;
  for (int i = tid; i < nb; i += TPB) lh[i] = 0;
  __syncthreads();

  const size_t  base = (size_t)row * N_COLS + (size_t)blk * CHUNK;
  const float4* p4   = (const float4*)(pred + base);
  const int4*   l4   = (const int4*)(label + base);
  const unsigned pref = s_pref;
  for (int i = tid; i < CHUNK / 4; i += TPB) {
    float4 v = p4[i]; int4 lbv = l4[i];
    float xs[4] = {v.x, v.y, v.z, v.w};
    int   ls[4] = {lbv.x, lbv.y, lbv.z, lbv.w};
#pragma unroll
    for (int j = 0; j < 4; ++j) {
      if (ls[j] == 0) {
        unsigned kb = __float_as_uint(xs[j]);
        if (ROUND == 2) { if ((kb >> 21) == pref) atomicAdd(&lh[(kb >> 10) & (NB12 - 1)], 1); }
        else            { if ((kb >> 10) == pref) atomicAdd(&lh[kb & (NB3 - 1)], 1); }
      }
    }
  }
  __syncthreads();
  int* gh = hist + row * NB12;
  for (int i = tid; i < nb; i += TPB) {
    int c = lh[i];
    if (c) atomicAdd(&gh[i], c);
  }
}

__global__ void k_scan2(Row* __restrict__ st, int* __restrict__ hist) {
  const int row = threadIdx.x;
  if (row >= B_ROWS) return;
  Row& r = st[row];
  if (r.done) return;
  int* h = hist + row * NB12;
  int rem = r.remaining, cum = 0, b = NB12 - 1;
  for (; b > 0; --b) { int c = h[b]; if (cum + c >= rem) break; cum += c; }
  r.prefix    = (r.prefix << 11) | (unsigned)b;
  r.remaining = rem - cum;
  for (int i = 0; i < NB12; ++i) h[i] = 0;
}

__global__ void k_scan3(Row* __restrict__ st, int* __restrict__ hist) {
  const int row = threadIdx.x;
  if (row >= B_ROWS) return;
  Row& r = st[row];
  if (r.done) return;
  int* h = hist + row * NB12;
  int rem = r.remaining, cum = 0, b = NB3 - 1;
  for (; b > 0; --b) { int c = h[b]; if (cum + c >= rem) break; cum += c; }
  unsigned vbits = (r.prefix << 10) | (unsigned)b;  // exact 32-bit pattern
  r.v_thr = __uint_as_float(vbits);
  r.r_eq  = rem - cum;                              // tie count (>=1)
  r.done  = 1;
}

// -------------------- Pass 4: weighted softmax sums ------------------------
__global__ void k_weighted(const float* __restrict__ pred,
                           const int*  __restrict__ label,
                           Row* __restrict__ st) {
  __shared__ float spm[CHUNK];       // 8 KB async-staged pred tile
  __shared__ float redS[TPB], redT[TPB];
  __shared__ int   s_skip;
  __shared__ float s_thr, s_max;
  const int row = blockIdx.y, blk = blockIdx.x, tid = threadIdx.x;
  if (tid == 0) {
    s_skip = (st[row].k_eff <= 0);
    s_thr  = st[row].v_thr;
    s_max  = __uint_as_float(st[row].neg_max_bits);
  }
  __syncthreads();
  if (s_skip) return;

  const size_t base = (size_t)row * N_COLS + (size_t)blk * CHUNK;

  // async pred copy overlapped with label fetch into registers
  async_copy_tile(pred + base, spm);
  const int4* l4 = (const int4*)(label + base);
  int4 lb[GPT];
#pragma unroll
  for (int j = 0; j < GPT; ++j) lb[j] = l4[j * TPB + tid];
  async_wait_all();
  __syncthreads();

  float S = 0.f, T = 0.f;
  const float thr = s_thr, M = s_max;
#pragma unroll
  for (int j = 0; j < GPT; ++j) {
    const int e = (j * TPB + tid) * 4;
    int ls[4] = {lb[j].x, lb[j].y, lb[j].z, lb[j].w};
#pragma unroll
    for (int c = 0; c < 4; ++c) {
      float x = spm[e + c];
      if (ls[c] == 0 && x > thr) {
        float ev = expf(x - M);     // x <= M, no overflow
        S += ev; T += x * ev;
      }
    }
  }
  redS[tid] = S; redT[tid] = T;
  __syncthreads();
  for (int s = TPB / 2; s > 0; s >>= 1) {
    if (tid < s) { redS[tid] += redS[tid + s]; redT[tid] += redT[tid + s]; }
    __syncthreads();
  }
  if (tid == 0) {
    atomicAdd(&st[row].S, redS[0]);
    atomicAdd(&st[row].T, redT[0]);
  }
}

// -------------------- Finalize: per-row loss + valid mean ------------------
__global__ void k_finalize(Row* __restrict__ st, float* __restrict__ out) {
  __shared__ float ssum[B_ROWS];
  __shared__ int   scnt[B_ROWS];
  const int row = threadIdx.x;
  float loss = 0.f; int valid = 0;
  if (row < B_ROWS) {
    Row r = st[row];
    valid = (r.hard > 0);
    const float M = __uint_as_float(r.neg_max_bits);
    float S = r.S, T = r.T;
    if (r.r_eq > 0) {                       // tied threshold entries (exact)
      float e = expf(r.v_thr - M);
      S += (float)r.r_eq * e;
      T += (float)r.r_eq * r.v_thr * e;
    }
    float neg_dist = (r.num_neg > 0 && S > 0.f) ? (T / S) : -1e30f;
    float base = (r.num_pos > 0)
                   ? (r.sum_pos / (float)(r.num_pos > 1 ? r.num_pos : 1))
                   : 1.0f;
    float z  = LOSS_L * (neg_dist - base + MARGIN);
    float sp = (z > 15.f) ? z : log1pf(expf(z));   // softplus
    loss = sp / LOSS_L;
  }
  ssum[row] = valid ? loss : 0.f;
  scnt[row] = valid;
  __syncthreads();
  for (int s = B_ROWS / 2; s > 0; s >>= 1) {
    if (row < s) { ssum[row] += ssum[row + s]; scnt[row] += scnt[row + s]; }
    __syncthreads();
  }
  if (row == 0) out[0] = (scnt[0] > 0) ? (ssum[0] / (float)scnt[0]) : 0.0f;
}

// ---------------------------------------------------------------------------
extern "C" void kernel_launch(void* const* d_in, const int* in_sizes, int n_in,
                              void* d_out, int out_size, void* d_ws, size_t ws_size,
                              hipStream_t stream) {
  const float* pred  = (const float*)d_in[0];
  const int*   label = (const int*)d_in[1];
  float*       out   = (float*)d_out;

  int* hist = (int*)d_ws;                                         // 1 MB
  Row* st   = (Row*)((char*)d_ws + (size_t)B_ROWS * NB12 * sizeof(int));
  const size_t clr = (size_t)B_ROWS * NB12 * sizeof(int) + (size_t)B_ROWS * sizeof(Row);
  hipMemsetAsync(d_ws, 0, clr, stream);                           // capture-safe

  dim3 grid(BPR, B_ROWS), blk(TPB);
  k_stats_hist1<<<grid, blk, 0, stream>>>(pred, label, st, hist);
  k_scan1<<<1, B_ROWS, 0, stream>>>(st, hist);
  k_refine<2><<<grid, blk, 0, stream>>>(pred, label, st, hist);
  k_scan2<<<1, B_ROWS, 0, stream>>>(st, hist);
  k_refine<3><<<grid, blk, 0, stream>>>(pred, label, st, hist);
  k_scan3<<<1, B_ROWS, 0, stream>>>(st, hist);
  k_weighted<<<grid, blk, 0, stream>>>(pred, label, st);
  k_finalize<<<1, B_ROWS, 0, stream>>>(st, out);
}